// SoftSkeletonize_44092134261330
// MI455X (gfx1250) — compile-verified
//
#include <hip/hip_runtime.h>

// SoftSkeletonize for (B=2, C=1, D=128, H=256, W=256) fp32.
// soft_erode == 3x3x3 min-pool (the (3,3,1) window is a subset of (3,3,3)).
// Memory-bound stencil: CDNA5 async global->LDS loads + split wait counters,
// float4-vectorized elementwise passes, fused open = dilate(erode(.)).

#define BN 2
#define DN 128
#define HN 256
#define WN 256
#define NTOT (BN * DN * HN * WN)   // 16,777,216
#define NV4  (NTOT / 4)            //  4,194,304 float4

#define TD 4
#define TH 8
#define TW 32

// ---------------- CDNA5 async global->LDS helpers ----------------
__device__ __forceinline__ void async_ld_b32(const float* g, const float* lds_generic) {
    unsigned loff = (unsigned)(unsigned long long)(uintptr_t)lds_generic; // low 32 bits = LDS offset
    asm volatile("global_load_async_to_lds_b32 %0, %1, off"
                 :: "v"(loff), "v"(g) : "memory");
}
__device__ __forceinline__ void wait_async0() {
    asm volatile("s_wait_asynccnt 0" ::: "memory");
}

__device__ __forceinline__ float min3f(float a, float b, float c) { return fminf(a, fminf(b, c)); }
__device__ __forceinline__ float max3f(float a, float b, float c) { return fmaxf(a, fmaxf(b, c)); }
__device__ __forceinline__ int clampi(int v, int hi) { return v < 0 ? 0 : (v > hi ? hi : v); }
__device__ __forceinline__ float leaky(float x) { return x > 0.f ? x : 0.01f * x; }

// ---------------- erode: 3x3x3 min pool, clamp-to-edge (== +inf SAME pad) ----
__global__ __launch_bounds__(256) void erode3(const float* __restrict__ src,
                                              float* __restrict__ dst) {
    __shared__ float sA[6 * 10 * 36];
    __shared__ float sB[6 * 10 * 36];
    const int tid = threadIdx.x;
    const int w0 = blockIdx.x * TW;
    const int h0 = blockIdx.y * TH;
    const int b  = blockIdx.z / (DN / TD);
    const int d0 = (blockIdx.z % (DN / TD)) * TD;
    const size_t base = (size_t)b * DN * HN * WN;

    // halo region 6 x 10 x 34 (pitch 36), async loads, clamp-to-edge addressing
    for (int i = tid; i < 6 * 10 * 34; i += 256) {
        int ld = i / 340, r = i % 340, lh = r / 34, lw = r % 34;
        int gd = clampi(d0 + ld - 1, DN - 1);
        int gh = clampi(h0 + lh - 1, HN - 1);
        int gw = clampi(w0 + lw - 1, WN - 1);
        async_ld_b32(&src[base + ((size_t)gd * HN + gh) * WN + gw],
                     &sA[(ld * 10 + lh) * 36 + lw]);
    }
    wait_async0();
    __syncthreads();
    // W-min: 6 x 10 x 32 -> sB
    for (int i = tid; i < 6 * 10 * 32; i += 256) {
        int ld = i / 320, r = i % 320, lh = r / 32, wo = r % 32;
        const float* p = &sA[(ld * 10 + lh) * 36 + wo];
        sB[(ld * 10 + lh) * 36 + wo] = min3f(p[0], p[1], p[2]);
    }
    __syncthreads();
    // H-min: 6 x 8 x 32 -> sA
    for (int i = tid; i < 6 * 8 * 32; i += 256) {
        int ld = i / 256, r = i % 256, ho = r / 32, wo = r % 32;
        sA[(ld * 10 + ho) * 36 + wo] = min3f(sB[(ld * 10 + ho) * 36 + wo],
                                             sB[(ld * 10 + ho + 1) * 36 + wo],
                                             sB[(ld * 10 + ho + 2) * 36 + wo]);
    }
    __syncthreads();
    // D-min + store: 4 x 8 x 32
    for (int i = tid; i < TD * TH * TW; i += 256) {
        int dd = i / 256, r = i % 256, ho = r / 32, wo = r % 32;
        float v = min3f(sA[(dd * 10 + ho) * 36 + wo],
                        sA[((dd + 1) * 10 + ho) * 36 + wo],
                        sA[((dd + 2) * 10 + ho) * 36 + wo]);
        dst[base + ((size_t)(d0 + dd) * HN + (h0 + ho)) * WN + (w0 + wo)] = v;
    }
}

// ---------------- open: dilate(erode(x)) fused, halo 2 -----------------------
__global__ __launch_bounds__(256) void open3(const float* __restrict__ src,
                                             float* __restrict__ dst) {
    __shared__ float sA[8 * 12 * 36];
    __shared__ float sB[8 * 12 * 36];
    const int tid = threadIdx.x;
    const int w0 = blockIdx.x * TW;
    const int h0 = blockIdx.y * TH;
    const int b  = blockIdx.z / (DN / TD);
    const int d0 = (blockIdx.z % (DN / TD)) * TD;
    const size_t base = (size_t)b * DN * HN * WN;

    // halo region 8 x 12 x 36 (pitch 36 exact), async loads
    for (int i = tid; i < 8 * 12 * 36; i += 256) {
        int ld = i / 432, r = i % 432, lh = r / 36, lw = r % 36;
        int gd = clampi(d0 + ld - 2, DN - 1);
        int gh = clampi(h0 + lh - 2, HN - 1);
        int gw = clampi(w0 + lw - 2, WN - 1);
        async_ld_b32(&src[base + ((size_t)gd * HN + gh) * WN + gw],
                     &sA[(ld * 12 + lh) * 36 + lw]);
    }
    wait_async0();
    __syncthreads();
    // W-min: 8 x 12 x 34 -> sB   (wo maps to w = w0+wo-1)
    for (int i = tid; i < 8 * 12 * 34; i += 256) {
        int ld = i / 408, r = i % 408, lh = r / 34, wo = r % 34;
        const float* p = &sA[(ld * 12 + lh) * 36 + wo];
        sB[(ld * 12 + lh) * 36 + wo] = min3f(p[0], p[1], p[2]);
    }
    __syncthreads();
    // H-min: 8 x 10 x 34 -> sA   (ho maps to h = h0+ho-1)
    for (int i = tid; i < 8 * 10 * 34; i += 256) {
        int ld = i / 340, r = i % 340, ho = r / 34, wo = r % 34;
        sA[(ld * 12 + ho) * 36 + wo] = min3f(sB[(ld * 12 + ho) * 36 + wo],
                                             sB[(ld * 12 + ho + 1) * 36 + wo],
                                             sB[(ld * 12 + ho + 2) * 36 + wo]);
    }
    __syncthreads();
    // D-min: 6 x 10 x 34 -> sB   (dd maps to d = d0+dd-1)  == eroded
    for (int i = tid; i < 6 * 10 * 34; i += 256) {
        int dd = i / 340, r = i % 340, ho = r / 34, wo = r % 34;
        sB[(dd * 12 + ho) * 36 + wo] = min3f(sA[(dd * 12 + ho) * 36 + wo],
                                             sA[((dd + 1) * 12 + ho) * 36 + wo],
                                             sA[((dd + 2) * 12 + ho) * 36 + wo]);
    }
    __syncthreads();
    // W-max: 6 x 10 x 32 -> sA   (wo maps to w = w0+wo)
    for (int i = tid; i < 6 * 10 * 32; i += 256) {
        int dd = i / 320, r = i % 320, ho = r / 32, wo = r % 32;
        const float* p = &sB[(dd * 12 + ho) * 36 + wo];
        sA[(dd * 12 + ho) * 36 + wo] = max3f(p[0], p[1], p[2]);
    }
    __syncthreads();
    // H-max: 6 x 8 x 32 -> sB    (ho maps to h = h0+ho)
    for (int i = tid; i < 6 * 8 * 32; i += 256) {
        int dd = i / 256, r = i % 256, ho = r / 32, wo = r % 32;
        sB[(dd * 12 + ho) * 36 + wo] = max3f(sA[(dd * 12 + ho) * 36 + wo],
                                             sA[(dd * 12 + ho + 1) * 36 + wo],
                                             sA[(dd * 12 + ho + 2) * 36 + wo]);
    }
    __syncthreads();
    // D-max + store: 4 x 8 x 32
    for (int i = tid; i < TD * TH * TW; i += 256) {
        int dd = i / 256, r = i % 256, ho = r / 32, wo = r % 32;
        float v = max3f(sB[(dd * 12 + ho) * 36 + wo],
                        sB[((dd + 1) * 12 + ho) * 36 + wo],
                        sB[((dd + 2) * 12 + ho) * 36 + wo]);
        dst[base + ((size_t)(d0 + dd) * HN + (h0 + ho)) * WN + (w0 + wo)] = v;
    }
}

// ---------------- init: skel = leaky(img - open0); prev = 0; imgState = img --
// float4-vectorized: 4096 blocks x 256 threads x 4 float4 = NTOT elements
__global__ __launch_bounds__(256) void init_k(const float4* __restrict__ img,
                                              const float4* __restrict__ open0,
                                              float4* __restrict__ skel,
                                              float4* __restrict__ prev,
                                              float4* __restrict__ imgState,
                                              int* flag) {
    size_t i0 = (size_t)blockIdx.x * 1024 + threadIdx.x;
    const float4 z = make_float4(0.f, 0.f, 0.f, 0.f);
    #pragma unroll
    for (int k = 0; k < 4; ++k) {
        size_t i = i0 + (size_t)k * 256;
        float4 x = img[i], o = open0[i];
        float4 s;
        s.x = leaky(x.x - o.x);
        s.y = leaky(x.y - o.y);
        s.z = leaky(x.z - o.z);
        s.w = leaky(x.w - o.w);
        skel[i] = s;
        prev[i] = z;
        imgState[i] = x;
    }
    if (blockIdx.x == 0 && threadIdx.x == 0) *flag = 1;
}

// ---------------- fused elementwise update + partial reduction ---------------
__device__ __forceinline__ float upd1(float x2, float o2, float s, float p,
                                      float* s2_out) {
    float dl = leaky(x2 - o2);
    float u  = leaky(dl - s * dl);
    float s2 = s + u;
    *s2_out = s2;
    return fabsf(s2 - p);
}

__global__ __launch_bounds__(256) void update_k(const float4* __restrict__ img2,
                                                const float4* __restrict__ open2,
                                                float4* __restrict__ skel,
                                                float4* __restrict__ prev,
                                                float4* __restrict__ imgState,
                                                float* __restrict__ partials,
                                                const int* flag) {
    __shared__ float red[256];
    const int act = *flag;
    size_t i0 = (size_t)blockIdx.x * 1024 + threadIdx.x;
    float lsum = 0.f;
    #pragma unroll
    for (int k = 0; k < 4; ++k) {
        size_t i = i0 + (size_t)k * 256;
        float4 x2 = img2[i], o2 = open2[i], s = skel[i], p = prev[i];
        float4 s2;
        lsum += upd1(x2.x, o2.x, s.x, p.x, &s2.x);
        lsum += upd1(x2.y, o2.y, s.y, p.y, &s2.y);
        lsum += upd1(x2.z, o2.z, s.z, p.z, &s2.z);
        lsum += upd1(x2.w, o2.w, s.w, p.w, &s2.w);
        if (act) { skel[i] = s2; prev[i] = s2; imgState[i] = x2; }
    }
    red[threadIdx.x] = lsum;
    __syncthreads();
    for (int off = 128; off > 0; off >>= 1) {
        if (threadIdx.x < off) red[threadIdx.x] += red[threadIdx.x + off];
        __syncthreads();
    }
    if (threadIdx.x == 0) partials[blockIdx.x] = red[0];
}

// ---------------- fold partials, update active flag (deterministic order) ----
__global__ __launch_bounds__(256) void reduce_k(const float* __restrict__ partials,
                                                int* flag) {
    __shared__ float red[256];
    float s = 0.f;
    for (int k = threadIdx.x; k < 4096; k += 256) s += partials[k];
    red[threadIdx.x] = s;
    __syncthreads();
    for (int off = 128; off > 0; off >>= 1) {
        if (threadIdx.x < off) red[threadIdx.x] += red[threadIdx.x + off];
        __syncthreads();
    }
    if (threadIdx.x == 0) {
        float dn = red[0] / (float)NTOT;
        *flag = (*flag != 0 && dn >= 1e-4f) ? 1 : 0;
    }
}

extern "C" void kernel_launch(void* const* d_in, const int* in_sizes, int n_in,
                              void* d_out, int out_size, void* d_ws, size_t ws_size,
                              hipStream_t stream) {
    const float* img = (const float*)d_in[0];
    float* skel = (float*)d_out;

    float* ws = (float*)d_ws;
    float* imgState = ws;
    float* prev = ws + (size_t)NTOT;
    float* tA   = ws + 2 * (size_t)NTOT;
    float* tB   = ws + 3 * (size_t)NTOT;
    float* partials = ws + 4 * (size_t)NTOT;
    int* flag = (int*)(partials + 4096);

    dim3 pg(WN / TW, HN / TH, BN * (DN / TD));
    dim3 pb(256);

    // skel0 = leaky(img - open(img)); prev = 0; imgState = img; active = 1
    open3<<<pg, pb, 0, stream>>>(img, tB);
    init_k<<<4096, 256, 0, stream>>>((const float4*)img, (const float4*)tB,
                                     (float4*)skel, (float4*)prev,
                                     (float4*)imgState, flag);

    for (int it = 0; it < 20; ++it) {
        erode3<<<pg, pb, 0, stream>>>(imgState, tA);      // img2 = erode(img)
        open3<<<pg, pb, 0, stream>>>(tA, tB);             // open2 = dilate(erode(img2))
        update_k<<<4096, 256, 0, stream>>>((const float4*)tA, (const float4*)tB,
                                           (float4*)skel, (float4*)prev,
                                           (float4*)imgState, partials, flag);
        reduce_k<<<1, 256, 0, stream>>>(partials, flag);  // dn + early-stop flag
    }
}